// MultiEmbedding_86294482911710
// MI455X (gfx1250) — compile-verified
//
#include <hip/hip_runtime.h>
#include <hip/hip_bf16.h>
#include <stdint.h>

// ---------------------------------------------------------------------------
// MultiEmbedding: out[n,s,:] = sum_l weight[l, x[n,l,s], :]
// Memory-bound gather+sum. CDNA5 path: TDM gather-mode tensor_load_to_lds
// (one descriptor = 16 gathered rows = 8 levels x 2 output positions),
// double-buffered in LDS, read back via ds_load_b128 (address_space(3)),
// NT b128 stores for the 64 MiB output stream.
// ---------------------------------------------------------------------------

typedef unsigned int u32;
typedef u32   v4u __attribute__((ext_vector_type(4)));
typedef int   v4i __attribute__((ext_vector_type(4)));
typedef int   v8i __attribute__((ext_vector_type(8)));
typedef float v4f __attribute__((ext_vector_type(4)));

typedef __attribute__((address_space(3))) const v4f* lds_cv4f;

// Width-correct integer->LDS-pointer cast for both compilation passes:
// device AS(3) pointers are 32-bit, host pointers are 64-bit.
__device__ __host__ __forceinline__ lds_cv4f lds_ptr_from_off(u32 off) {
#if defined(__HIP_DEVICE_COMPILE__)
  return (lds_cv4f)off;
#else
  return (lds_cv4f)(uintptr_t)off;  // host pass: never executed, just typed
#endif
}

constexpr int kL = 8;      // levels
constexpr int kK = 1024;   // tokens per level
constexpr int kD = 512;    // token dim (floats)
constexpr int kN = 8;      // batch
constexpr int kS = 4096;   // seq
constexpr int kRowsPerBlock = 16;            // output (n,s) rows per block
constexpr int kIters = kRowsPerBlock / 2;    // 2 rows per TDM gather

// Issue one TDM gather: 16 rows (l*1024 + x[n,l,s]) of 512 f32 each -> LDS.
// Must be executed by exactly one wave (wave 0 of the block).
__device__ __forceinline__ void issue_gather_tdm(const int* __restrict__ x,
                                                 const float* __restrict__ w,
                                                 int n, int sA, u32 ldsOff) {
  const unsigned lane = threadIdx.x & 31u;
  u32 rowidx = 0u;
  if (lane < 16u) {
    const int l = (int)(lane & 7u);   // level
    const int g = (int)(lane >> 3);   // which of the 2 output positions
    const int k = x[((size_t)n * kL + l) * kS + (sA + g)];
    rowidx = (u32)(l * kK + k);       // combined row id, < 8192, fits 16 bits
  }
  // Pack 16 x 16-bit row indices into 8 scalar words (D# groups 2+3).
  u32 pk[8];
#pragma unroll
  for (int p = 0; p < 8; ++p) {
    const u32 lo = (u32)__builtin_amdgcn_readlane(rowidx, 2 * p);
    const u32 hi = (u32)__builtin_amdgcn_readlane(rowidx, 2 * p + 1);
    pk[p] = lo | (hi << 16);
  }

  const uint64_t ga = (uint64_t)(uintptr_t)w;

  // D# group 0 (ISA 8.3): count=1, gather_mode=1 (16-bit indices),
  // lds_addr, global_addr[56:0], type=2.
  const v4u g0 = {
      0x80000001u,
      ldsOff,
      (u32)ga,
      (u32)((ga >> 32) & 0x01FFFFFFull) | (2u << 30)
  };
  // D# group 1 (ISA 8.4): data_size=2 (4B), tensor_dim0=512,
  // tensor_dim1=8192 rows, tile_dim0=512, tile_dim1=16 valid indices,
  // tensor_dim0_stride=512 elements. No multicast, no padding, no barrier.
  const v8i g1 = {
      (int)0x00020000u,            // [17:16] data_size = 4 bytes
      (int)((u32)kD << 16),        // [63:48]  tensor_dim0 lo
      (int)((u32)(kL * kK) << 16), // [95:80]  tensor_dim1 lo
      (int)((u32)kD << 16),        // [127:112] tile_dim0
      16,                          // [143:128] tile_dim1 = #indices
      kD,                          // [191:160] tensor_dim0_stride lo
      0, 0
  };
  // D# groups 2+3 (gather mode): row_index_0..15, 16-bit packed.
  const v4i g2 = { (int)pk[0], (int)pk[1], (int)pk[2], (int)pk[3] };
  const v4i g3 = { (int)pk[4], (int)pk[5], (int)pk[6], (int)pk[7] };

#if __clang_major__ >= 23
  const v8i gx = {0, 0, 0, 0, 0, 0, 0, 0};
  __builtin_amdgcn_tensor_load_to_lds(g0, g1, g2, g3, gx, 0);
#else
  __builtin_amdgcn_tensor_load_to_lds(g0, g1, g2, g3, 0);
#endif
}

__global__ __launch_bounds__(256) void MultiEmbedding_tdm_gather_kernel(
    const int* __restrict__ x, const float* __restrict__ w,
    float* __restrict__ out) {
  // Double buffer: 2 x (16 rows x 512 f32) = 64 KiB LDS, 128B-aligned tiles.
  __shared__ __align__(128) float buf[2][16 * kD];

  const int row0   = (int)blockIdx.x * kRowsPerBlock;  // flattened n*S+s
  const int n      = row0 / kS;
  const int s_base = row0 % kS;
  const unsigned tid = threadIdx.x;
  const unsigned wid = tid >> 5;
  const unsigned g   = tid >> 7;    // 0..1 : which output row of the pair
  const unsigned dd  = tid & 127u;  // float4 index within the 512-f32 row

  // 32-bit LDS byte offsets of the two buffers (generic addr low 32 bits).
  const u32 off0 = (u32)(uintptr_t)(void*)&buf[0][0];
  const u32 off1 = (u32)(uintptr_t)(void*)&buf[1][0];

  // Prologue: DMA for iteration 0.
  if (wid == 0) {
    issue_gather_tdm(x, w, n, s_base, off0);
  }

  for (int it = 0; it < kIters; ++it) {
    if (wid == 0) {
      if (it + 1 < kIters) {
        // Overlap: issue next gather, then wait only for the current one
        // (TENSORcnt completes in order per wave).
        issue_gather_tdm(x, w, n, s_base + 2 * (it + 1),
                         ((it + 1) & 1) ? off1 : off0);
        __builtin_amdgcn_s_wait_tensorcnt(1);
      } else {
        __builtin_amdgcn_s_wait_tensorcnt(0);
      }
    }
    __syncthreads();  // DMA(it) complete -> safe to read buf[it&1]

    // Launder the 32-bit LDS offset: the TDM wrote LDS behind the compiler's
    // back. Reading through an AS(3) pointer guarantees ds_load_b128.
    u32 off = (it & 1) ? off1 : off0;
    asm volatile("" : "+s"(off)::"memory");

    lds_cv4f r = lds_ptr_from_off(off) + (g * 8) * (kD / 4) + dd;
    v4f acc = {0.f, 0.f, 0.f, 0.f};
#pragma unroll
    for (int l = 0; l < kL; ++l) acc += r[l * (kD / 4)];  // ds_load_b128

    const int s = s_base + 2 * it + (int)g;
    v4f* o = (v4f*)out + ((size_t)n * kS + s) * (kD / 4) + dd;
    // Output (64 MiB) is write-once: NT keeps the weight tables hot in L2.
    __builtin_nontemporal_store(acc, o);

    __syncthreads();  // reads done -> DMA(it+2) may overwrite buf[it&1]
  }
}

extern "C" void kernel_launch(void* const* d_in, const int* in_sizes, int n_in,
                              void* d_out, int out_size, void* d_ws,
                              size_t ws_size, hipStream_t stream) {
  // setup_inputs order: x (int), weight (f32). Output f32 [N,S,D].
  const int*   x = (const int*)d_in[0];
  const float* w = (const float*)d_in[1];
  float*     out = (float*)d_out;

  const int total_rows = kN * kS;                    // 32768
  dim3 grid(total_rows / kRowsPerBlock);             // 2048 blocks
  dim3 block(256);                                   // 8 waves (wave32)
  MultiEmbedding_tdm_gather_kernel<<<grid, block, 0, stream>>>(x, w, out);
}